// MMMambaEncoderLayer_40097814675762
// MI455X (gfx1250) — compile-verified
//
#include <hip/hip_runtime.h>
#include <math.h>

// ---------------------------------------------------------------------------
// BiMamba encoder layer for gfx1250 (MI455X).
// fp32 end-to-end; matrix ops via V_WMMA_F32_16X16X4_F32.
// ---------------------------------------------------------------------------

#define D_MODEL 256
#define D_INNER 512
#define D_STATE 16
#define D_CONV  4
#define DT_RANK 16
#define BATCH   8
#define SEQLEN  1024
#define ROWS    (BATCH * SEQLEN)      // 8192
#define LN_EPS  1e-6f

typedef float v2f __attribute__((ext_vector_type(2)));
typedef float v4f __attribute__((ext_vector_type(4)));
typedef float v8f __attribute__((ext_vector_type(8)));

// ---------------------------------------------------------------------------
// WMMA wrapper: D = A(16x4,f32) * B(4x16,f32) + C(16x16,f32)
// A frag: lane L -> M = L%16, v[j] = A[M][2*(L/16)+j]
// B frag: lane L -> N = L%16, v[j] = B[2*(L/16)+j][N]
// C/D    : lane L, vgpr r -> C[r + 8*(L/16)][L%16]
// ---------------------------------------------------------------------------
__device__ inline v8f wmma_f32_16x16x4(v2f a, v2f b, v8f c) {
  return __builtin_amdgcn_wmma_f32_16x16x4_f32(false, a, false, b, (short)0, c,
                                               false, false);
}

__device__ inline float silu_f(float x) { return x / (1.f + __expf(-x)); }
__device__ inline float softplus_f(float x) {
  return fmaxf(x, 0.f) + log1pf(__expf(-fabsf(x)));
}

// ---------------------------------------------------------------------------
// GEMM: C[M,N] = A[M,K] * W[N,K]^T  (+ epilogue)
//   EP = 0: plain store
//   EP = 1: softplus(acc + bias[n])    (dt_proj)
//   KFULL: K % 32 == 0  -> unguarded float4 staging (b128 loads/stores)
//   NFULL: N % 64 == 0  -> no N guards anywhere
// Block: 256 threads (8 waves), tile 64(M) x 64(N) x 32(K), double-buffered
// LDS staging. Wave layout: 4 along M (16 rows) x 2 along N (32 cols).
// ---------------------------------------------------------------------------
template <int EP, bool KFULL, bool NFULL>
__global__ __launch_bounds__(256) void gemm_f32_wmma(
    const float* __restrict__ A, const float* __restrict__ W,
    const float* __restrict__ bias, float* __restrict__ C,
    int M, int N, int K, int lda) {
  // pitch 36 floats: rows 16B-aligned (144B) for b128, even for v2f reads.
  __shared__ __align__(16) float As[2][64][36];
  __shared__ __align__(16) float Ws[2][64][36];

  const int tid = threadIdx.x;
  const int m0 = blockIdx.x * 64;
  const int n0 = blockIdx.y * 64;

  const int wid = tid >> 5;
  const int lane = tid & 31;
  const int wm = (wid & 3) * 16;       // wave row offset in tile
  const int wn = (wid >> 2) * 32;      // wave col offset in tile
  const int mrow = wm + (lane & 15);
  const int nrow = wn + (lane & 15);
  const int khalf = (lane >> 4) << 1;  // 0 or 2

  const int ldr = tid >> 2;            // 0..63 (staging row)
  const int kc = (tid & 3) * 8;        // 0,8,16,24

  v8f acc0 = {};
  v8f acc1 = {};

  auto stage = [&](int k0, int bufIdx) {
    if (KFULL) {
      const float* ap = &A[(size_t)(m0 + ldr) * lda + k0 + kc];
      v4f a0 = *(const v4f*)ap;
      v4f a1 = *(const v4f*)(ap + 4);
      v4f w0 = {}, w1 = {};
      if (NFULL || (n0 + ldr) < N) {
        const float* wp = &W[(size_t)(n0 + ldr) * K + k0 + kc];
        w0 = *(const v4f*)wp;
        w1 = *(const v4f*)(wp + 4);
      }
      if (k0 + 32 < K) {  // warm L2 one tile further ahead
        __builtin_prefetch(ap + 32, 0, 0);
        if (NFULL || (n0 + ldr) < N)
          __builtin_prefetch(&W[(size_t)(n0 + ldr) * K + k0 + 32 + kc], 0, 0);
      }
      *(v4f*)&As[bufIdx][ldr][kc] = a0;
      *(v4f*)&As[bufIdx][ldr][kc + 4] = a1;
      *(v4f*)&Ws[bufIdx][ldr][kc] = w0;
      *(v4f*)&Ws[bufIdx][ldr][kc + 4] = w1;
    } else {
#pragma unroll
      for (int i = 0; i < 8; ++i) {
        const int k = k0 + kc + i;
        As[bufIdx][ldr][kc + i] =
            (k < K) ? A[(size_t)(m0 + ldr) * lda + k] : 0.f;
        Ws[bufIdx][ldr][kc + i] =
            ((NFULL || (n0 + ldr) < N) && k < K)
                ? W[(size_t)(n0 + ldr) * K + k]
                : 0.f;
      }
    }
  };

  int buf = 0;
  stage(0, 0);
  __syncthreads();

  for (int k0 = 0; k0 < K; k0 += 32) {
    if (k0 + 32 < K) stage(k0 + 32, buf ^ 1);

#pragma unroll
    for (int kk = 0; kk < 32; kk += 4) {
      const int kf = kk + khalf;
      v2f a  = *(const v2f*)&As[buf][mrow][kf];
      v2f b0 = *(const v2f*)&Ws[buf][nrow][kf];
      v2f b1 = *(const v2f*)&Ws[buf][nrow + 16][kf];
      acc0 = wmma_f32_16x16x4(a, b0, acc0);
      acc1 = wmma_f32_16x16x4(a, b1, acc1);
    }
    __syncthreads();
    buf ^= 1;
  }

  // store C fragments
  const int rbase = m0 + wm + ((lane >> 4) << 3);
  const int c0 = n0 + wn + (lane & 15);
  const int c1 = c0 + 16;
#pragma unroll
  for (int r = 0; r < 8; ++r) {
    const int row = rbase + r;
    if (NFULL || c0 < N) {
      float v = acc0[r];
      if (EP == 1) v = softplus_f(v + bias[c0]);
      C[(size_t)row * N + c0] = v;
    }
    if (NFULL || c1 < N) {
      float v = acc1[r];
      if (EP == 1) v = softplus_f(v + bias[c1]);
      C[(size_t)row * N + c1] = v;
    }
  }
}

// ---------------------------------------------------------------------------
// Depthwise causal conv (width 4) + bias + SiLU.
// BWD=1 computes in flipped sequence space: xc[l'] = conv(x reversed)[l'].
// xz row stride is 2*D_INNER (xi occupies cols [0,512)).
// ---------------------------------------------------------------------------
template <int BWD>
__global__ __launch_bounds__(256) void conv_silu_kernel(
    const float* __restrict__ xz, const float* __restrict__ w,
    const float* __restrict__ cb, float* __restrict__ xc) {
  const int idx = blockIdx.x * 256 + threadIdx.x;           // < B*L*512
  const int d = idx & (D_INNER - 1);
  const int l = (idx >> 9) & (SEQLEN - 1);
  const int b = idx >> 19;

  float acc = cb[d];
#pragma unroll
  for (int k = 0; k < D_CONV; ++k) {
    const int src = BWD ? (SEQLEN - 1 - l + (D_CONV - 1) - k)
                        : (l - (D_CONV - 1) + k);
    if (src >= 0 && src < SEQLEN)
      acc = fmaf(w[d * D_CONV + k],
                 xz[(size_t)(b * SEQLEN + src) * (2 * D_INNER) + d], acc);
  }
  xc[(size_t)(b * SEQLEN + l) * D_INNER + d] = silu_f(acc);
}

// ---------------------------------------------------------------------------
// Selective scan. One block per batch element, 512 threads = one per channel.
// h[16], A[16] live in VGPRs; shared B/C (32 floats per step) staged in LDS
// 32 steps at a time (one barrier pair per 32 steps, broadcast reads).
// BWD: results are written flipped back to original order, pre-multiplied by
// silu(z[original l]); ACCUM adds onto the forward pass result.
// ---------------------------------------------------------------------------
__global__ __launch_bounds__(512) void scan_kernel(
    const float* __restrict__ xc, const float* __restrict__ proj,
    const float* __restrict__ dt, const float* __restrict__ xz,
    const float* __restrict__ A_log, const float* __restrict__ Dp,
    float* __restrict__ y_sum, int backward, int accumulate) {
  const int b = blockIdx.x;
  const int d = threadIdx.x;

  float Aa[D_STATE];
#pragma unroll
  for (int s = 0; s < D_STATE; ++s) Aa[s] = -__expf(A_log[d * D_STATE + s]);
  const float Dv = Dp[d];

  float h[D_STATE];
#pragma unroll
  for (int s = 0; s < D_STATE; ++s) h[s] = 0.f;

  __shared__ float bc[32][32];  // [step][0..15 = B, 16..31 = C]

  for (int l0 = 0; l0 < SEQLEN; l0 += 32) {
    __syncthreads();
    for (int j = d; j < 1024; j += D_INNER) {
      const int step = j >> 5, col = j & 31;
      bc[step][col] =
          proj[(size_t)(b * SEQLEN + l0 + step) * (DT_RANK + 2 * D_STATE) +
               DT_RANK + col];
    }
    __syncthreads();

    for (int i = 0; i < 32; ++i) {
      const int l = l0 + i;
      const size_t base = (size_t)(b * SEQLEN + l) * D_INNER + d;
      const float dtv = dt[base];
      const float xv = xc[base];
      const float dx = dtv * xv;
      float y = 0.f;
#pragma unroll
      for (int s = 0; s < D_STATE; ++s) {
        const float e = __expf(dtv * Aa[s]);
        h[s] = fmaf(h[s], e, dx * bc[i][s]);
        y = fmaf(h[s], bc[i][D_STATE + s], y);
      }
      const int lo = backward ? (SEQLEN - 1 - l) : l;
      const float zv =
          xz[(size_t)(b * SEQLEN + lo) * (2 * D_INNER) + D_INNER + d];
      const float out = (y + xv * Dv) * silu_f(zv);
      const size_t oidx = (size_t)(b * SEQLEN + lo) * D_INNER + d;
      if (accumulate)
        y_sum[oidx] += out;
      else
        y_sum[oidx] = out;
    }
  }
}

// ---------------------------------------------------------------------------
// out = x + LayerNorm(t) * g + b   — one wave32 per row of 256.
// ---------------------------------------------------------------------------
__global__ __launch_bounds__(256) void ln_residual_kernel(
    const float* __restrict__ x, const float* __restrict__ t,
    const float* __restrict__ g, const float* __restrict__ bta,
    float* __restrict__ out) {
  const int lane = threadIdx.x & 31;
  const int row = blockIdx.x * 8 + (threadIdx.x >> 5);

  float v[8];
  float sum = 0.f, sq = 0.f;
#pragma unroll
  for (int j = 0; j < 8; ++j) {
    const int col = lane + j * 32;
    const float val = t[(size_t)row * D_MODEL + col];
    v[j] = val;
    sum += val;
    sq = fmaf(val, val, sq);
  }
#pragma unroll
  for (int off = 16; off >= 1; off >>= 1) {
    sum += __shfl_xor(sum, off, 32);
    sq += __shfl_xor(sq, off, 32);
  }
  const float mean = sum * (1.f / D_MODEL);
  const float var = sq * (1.f / D_MODEL) - mean * mean;
  const float rstd = rsqrtf(var + LN_EPS);
#pragma unroll
  for (int j = 0; j < 8; ++j) {
    const int col = lane + j * 32;
    out[(size_t)row * D_MODEL + col] =
        x[(size_t)row * D_MODEL + col] + (v[j] - mean) * rstd * g[col] +
        bta[col];
  }
}

// ---------------------------------------------------------------------------
// Host side
// ---------------------------------------------------------------------------
struct StreamParams {
  const float *A_log, *A_log_b, *D, *D_b, *conv_b, *conv_b_b, *conv_w,
      *conv_w_b, *dt_proj_b, *dt_proj_b_b, *dt_proj_w, *dt_proj_w_b,
      *in_proj_w, *ln_b, *ln_g, *out_proj_w, *x_proj_w, *x_proj_w_b;
};

// param index within an 18-entry block, by name in SORTED order, for the
// dict-INSERTION ordering (in_proj_w, out_proj_w, ln_g, ln_b, conv_w, conv_b,
// x_proj_w, dt_proj_w, dt_proj_b, A_log, D, then the _b group).
static const int kInsertionIdx[18] = {
    9,  /*A_log*/ 16, /*A_log_b*/ 10, /*D*/ 17, /*D_b*/
    5,  /*conv_b*/ 12, /*conv_b_b*/ 4, /*conv_w*/ 11, /*conv_w_b*/
    8,  /*dt_proj_b*/ 15, /*dt_proj_b_b*/ 7, /*dt_proj_w*/ 14, /*dt_proj_w_b*/
    0,  /*in_proj_w*/ 3, /*ln_b*/ 2, /*ln_g*/ 1, /*out_proj_w*/
    6,  /*x_proj_w*/ 13 /*x_proj_w_b*/};

static StreamParams load_params(void* const* d_in, int base, bool sorted) {
  const float* f[18];
  for (int i = 0; i < 18; ++i)
    f[i] = (const float*)d_in[base + (sorted ? i : kInsertionIdx[i])];
  StreamParams p;
  p.A_log = f[0];      p.A_log_b = f[1];
  p.D = f[2];          p.D_b = f[3];
  p.conv_b = f[4];     p.conv_b_b = f[5];
  p.conv_w = f[6];     p.conv_w_b = f[7];
  p.dt_proj_b = f[8];  p.dt_proj_b_b = f[9];
  p.dt_proj_w = f[10]; p.dt_proj_w_b = f[11];
  p.in_proj_w = f[12]; p.ln_b = f[13];
  p.ln_g = f[14];      p.out_proj_w = f[15];
  p.x_proj_w = f[16];  p.x_proj_w_b = f[17];
  return p;
}

extern "C" void kernel_launch(void* const* d_in, const int* in_sizes, int n_in,
                              void* d_out, int out_size, void* d_ws,
                              size_t ws_size, hipStream_t stream) {
  // -------- resolve input ordering --------
  int aX, vX, aP, vP;
  if (in_sizes[0] == ROWS * D_MODEL) {  // x tensors first (insertion top-level)
    aX = 0; vX = 1; aP = 2; vP = 20;
  } else {  // params first (sorted top-level: a_params, a_x, v_params, v_x)
    aP = 0; aX = 18; vP = 19; vX = 37;
  }
  const bool sortedP = (in_sizes[aP] == D_INNER * D_STATE);  // A_log first

  const float* a_x = (const float*)d_in[aX];
  const float* v_x = (const float*)d_in[vX];
  StreamParams ap = load_params(d_in, aP, sortedP);
  StreamParams vp = load_params(d_in, vP, sortedP);

  // -------- workspace layout (reused for both streams, ~129 MB) --------
  size_t off = 0;
  auto carve = [&](size_t bytes) {
    void* p = (char*)d_ws + off;
    off += (bytes + 255) & ~(size_t)255;
    return (float*)p;
  };
  float* xz     = carve((size_t)ROWS * 2 * D_INNER * 4);
  float* xc_f   = carve((size_t)ROWS * D_INNER * 4);
  float* xc_b   = carve((size_t)ROWS * D_INNER * 4);
  float* dt_f   = carve((size_t)ROWS * D_INNER * 4);
  float* dt_b   = carve((size_t)ROWS * D_INNER * 4);
  float* proj_f = carve((size_t)ROWS * (DT_RANK + 2 * D_STATE) * 4);
  float* proj_b = carve((size_t)ROWS * (DT_RANK + 2 * D_STATE) * 4);
  float* y_sum  = carve((size_t)ROWS * D_INNER * 4);
  float* mm_out = carve((size_t)ROWS * D_MODEL * 4);

  auto run_stream = [&](const float* x, const StreamParams& p, float* out) {
    const int NPROJ = DT_RANK + 2 * D_STATE;  // 48
    const dim3 blk(256);
    // 1) in_proj: xz = x @ W_in^T            (8192x256 @ 256x1024)
    gemm_f32_wmma<0, true, true>
        <<<dim3(ROWS / 64, (2 * D_INNER) / 64), blk, 0, stream>>>(
            x, p.in_proj_w, nullptr, xz, ROWS, 2 * D_INNER, D_MODEL, D_MODEL);
    // 2) conv + silu, both directions
    const int convBlocks = (BATCH * SEQLEN * D_INNER) / 256;
    conv_silu_kernel<0><<<convBlocks, blk, 0, stream>>>(xz, p.conv_w, p.conv_b,
                                                        xc_f);
    conv_silu_kernel<1><<<convBlocks, blk, 0, stream>>>(xz, p.conv_w_b,
                                                        p.conv_b_b, xc_b);
    // 3) x_proj: proj = xc @ Wx^T            (8192x512 @ 512x48)
    gemm_f32_wmma<0, true, false><<<dim3(ROWS / 64, 1), blk, 0, stream>>>(
        xc_f, p.x_proj_w, nullptr, proj_f, ROWS, NPROJ, D_INNER, D_INNER);
    gemm_f32_wmma<0, true, false><<<dim3(ROWS / 64, 1), blk, 0, stream>>>(
        xc_b, p.x_proj_w_b, nullptr, proj_b, ROWS, NPROJ, D_INNER, D_INNER);
    // 4) dt_proj + softplus: dt = sp(proj[:, :16] @ Wdt^T + b)   (K=16)
    gemm_f32_wmma<1, false, true>
        <<<dim3(ROWS / 64, D_INNER / 64), blk, 0, stream>>>(
            proj_f, p.dt_proj_w, p.dt_proj_b, dt_f, ROWS, D_INNER, DT_RANK,
            NPROJ);
    gemm_f32_wmma<1, false, true>
        <<<dim3(ROWS / 64, D_INNER / 64), blk, 0, stream>>>(
            proj_b, p.dt_proj_w_b, p.dt_proj_b_b, dt_b, ROWS, D_INNER, DT_RANK,
            NPROJ);
    // 5) selective scans (fwd writes, bwd accumulates flipped-back)
    scan_kernel<<<BATCH, 512, 0, stream>>>(xc_f, proj_f, dt_f, xz, p.A_log,
                                           p.D, y_sum, 0, 0);
    scan_kernel<<<BATCH, 512, 0, stream>>>(xc_b, proj_b, dt_b, xz, p.A_log_b,
                                           p.D_b, y_sum, 1, 1);
    // 6) out_proj: mm_out = y_sum @ W_out^T  (8192x512 @ 512x256)
    gemm_f32_wmma<0, true, true>
        <<<dim3(ROWS / 64, D_MODEL / 64), blk, 0, stream>>>(
            y_sum, p.out_proj_w, nullptr, mm_out, ROWS, D_MODEL, D_INNER,
            D_INNER);
    // 7) residual + layernorm
    ln_residual_kernel<<<ROWS / 8, blk, 0, stream>>>(x, mm_out, p.ln_g, p.ln_b,
                                                     out);
  };

  float* out = (float*)d_out;
  run_stream(a_x, ap, out);                             // a_out
  run_stream(v_x, vp, out + (size_t)ROWS * D_MODEL);    // v_out
}